// U_MLP_79156247265943
// MI455X (gfx1250) — compile-verified
//
#include <hip/hip_runtime.h>
#include <math.h>

typedef __attribute__((ext_vector_type(2))) float v2f;
typedef __attribute__((ext_vector_type(8))) float v8f;

#define NB    16
#define SEQL  2048
#define DIMD  512
#define NE    16
#define SUBH  128
#define KTOT  (SEQL*DIMD)        // 1,048,576

// ---- router tiling ----
#define RCHUNKS 128
#define RCHUNK  (KTOT/RCHUNKS)   // 8192 k per block

// ---- FFN tiling ----
#define TS 64                    // seq rows per workgroup
#define KC 64                    // phase-A K chunk of DIM
// LDS float offsets (total 34304 floats = 137216 bytes; 2 workgroups/WGP at 320KB)
#define XS_OFF   0               // [64][68]   phase A X tile
#define XS_STR   68
#define W1_OFF   4352            // [64][132]  phase A W1 chunk (row-major k,col)
#define W1_STR   132
#define W2_OFF   0               // [128][132] phase B W2 chunk (overlays XS+W1)
#define W2_STR   132
#define H0_OFF   16896           // [64][130]  hidden activations, expert slot 0
#define H1_OFF   25216           //            expert slot 1
#define H_STR    130
#define BS_OFF   33536           // [512] b2[e0]+b2[e1]
#define B1_OFF   34048           // [2][128] b1[e0], b1[e1]
#define LDS_FLOATS 34304

// ---------------------------------------------------------------------------
// Kernel 1: router partial sums.  block (chunk c, batch b), 256 threads.
// partials[b][c][e] written exactly once, fixed reduction order -> deterministic
// (no atomics anywhere, so graph replays are bit-identical).
// ---------------------------------------------------------------------------
__global__ void __launch_bounds__(256) router_partial(
    const float* __restrict__ x, const float* __restrict__ Wsw,
    float* __restrict__ partials)
{
    const int c = blockIdx.x;          // k-chunk
    const int b = blockIdx.y;          // batch sample
    const int t = threadIdx.x;
    const long k0 = (long)c * RCHUNK;

    float acc[NE];
#pragma unroll
    for (int e = 0; e < NE; ++e) acc[e] = 0.0f;

    for (int j = 0; j < RCHUNK / 256; ++j) {           // 32 iters
        const long k = k0 + (long)j * 256 + t;
        const float xv = x[(long)b * KTOT + k];        // coalesced
        const float4* wr = (const float4*)(Wsw + k * NE); // 64B per thread, coalesced
        float w[NE];
        *(float4*)&w[0]  = wr[0];
        *(float4*)&w[4]  = wr[1];
        *(float4*)&w[8]  = wr[2];
        *(float4*)&w[12] = wr[3];
#pragma unroll
        for (int e = 0; e < NE; ++e) acc[e] = fmaf(xv, w[e], acc[e]);
    }

    // deterministic wave32 tree reduce, then fixed-order cross-wave sum
    __shared__ float smem[8 * NE];
#pragma unroll
    for (int e = 0; e < NE; ++e) {
#pragma unroll
        for (int d = 16; d > 0; d >>= 1)
            acc[e] += __shfl_down(acc[e], d, 32);
    }
    if ((t & 31) == 0) {
#pragma unroll
        for (int e = 0; e < NE; ++e) smem[(t >> 5) * NE + e] = acc[e];
    }
    __syncthreads();
    if (t < NE) {
        float s = 0.0f;
#pragma unroll
        for (int w = 0; w < 8; ++w) s += smem[w * NE + t];
        partials[((long)b * RCHUNKS + c) * NE + t] = s;
    }
}

// ---------------------------------------------------------------------------
// Kernel 2: reduce partials (fixed order), add bsw, pick top-2 per sample.
// Softmax is monotonic, so top-k(probs) == top-k(logits).
// Tie-break: strict '>' keeps lowest index first, matching jax.lax.top_k.
// ---------------------------------------------------------------------------
__global__ void __launch_bounds__(256) reduce_top2(
    const float* __restrict__ partials, const float* __restrict__ bsw,
    int* __restrict__ top2)
{
    __shared__ float lg[NB * NE];
    const int t = threadIdx.x;
    const int b = t >> 4, e = t & 15;
    float s = bsw[e];
    for (int c = 0; c < RCHUNKS; ++c)
        s += partials[((long)b * RCHUNKS + c) * NE + e];
    lg[t] = s;
    __syncthreads();
    if (t < NB) {
        const float* L = lg + t * NE;
        int i0 = 0; float v0 = L[0];
#pragma unroll
        for (int i = 1; i < NE; ++i) if (L[i] > v0) { v0 = L[i]; i0 = i; }
        int i1 = 0; float v1 = -3.4e38f;
#pragma unroll
        for (int i = 0; i < NE; ++i)
            if (i != i0 && L[i] > v1) { v1 = L[i]; i1 = i; }
        top2[t * 2 + 0] = i0;
        top2[t * 2 + 1] = i1;
    }
}

// ---------------------------------------------------------------------------
// Kernel 3: dual-expert FFN with V_WMMA_F32_16X16X4_F32 (f32 in / f32 acc ->
// reference-level precision).  grid (SEQ/TS=32, NB=16), 256 threads = 8 waves.
// Wave w: 16-row strip (w>>1), 64-col half (w&1); 4 v8f accumulator tiles.
// ---------------------------------------------------------------------------
__global__ void __launch_bounds__(256) moe_ffn(
    const float* __restrict__ x,  const float* __restrict__ W1,
    const float* __restrict__ b1, const float* __restrict__ W2,
    const float* __restrict__ b2, const int* __restrict__ top2,
    float* __restrict__ out)
{
    extern __shared__ float lds[];
    const int b     = blockIdx.y;
    const int s0    = blockIdx.x * TS;
    const int t     = threadIdx.x;
    const int lane  = t & 31;
    const int w     = t >> 5;       // wave 0..7
    const int strip = w >> 1;       // 16-row strip 0..3
    const int half  = w & 1;        // 64-col half
    const int lh    = lane >> 4;    // lane group (K offset selector)
    const int lm    = lane & 15;

    const int e0 = top2[b * 2 + 0];
    const int e1 = top2[b * 2 + 1];

    // hoist small biases into LDS (read-only afterwards)
    for (int c = t; c < DIMD; c += 256)
        lds[BS_OFF + c] = b2[e0 * DIMD + c] + b2[e1 * DIMD + c];
    lds[B1_OFF + t] = b1[((t >= SUBH) ? e1 : e0) * SUBH + (t & (SUBH - 1))];

    // ---------------- Phase A: H_slot = gelu(X @ W1[e] + b1[e]) -------------
    for (int slot = 0; slot < 2; ++slot) {
        const int e = slot ? e1 : e0;
        const float* W1e = W1 + (long)e * DIMD * SUBH;
        float* H = lds + (slot ? H1_OFF : H0_OFF);
        v8f acc[4] = {};
        for (int kc = 0; kc < DIMD; kc += KC) {
            __syncthreads();
            // stage X tile [64 rows][64 k], stride 68 (16B aligned)
            for (int i = t; i < TS * KC / 4; i += 256) {     // 1024 float4
                const int row = i >> 4, q = i & 15;
                float4 v = *(const float4*)(x + ((long)(b * SEQL + s0 + row)) * DIMD + kc + q * 4);
                *(float4*)(lds + XS_OFF + row * XS_STR + q * 4) = v;
            }
            // stage W1 chunk [64 k][128 col], stride 132
            for (int i = t; i < KC * SUBH / 4; i += 256) {   // 2048 float4
                const int k = i >> 5, q = i & 31;
                float4 v = *(const float4*)(W1e + (long)(kc + k) * SUBH + q * 4);
                *(float4*)(lds + W1_OFF + k * W1_STR + q * 4) = v;
            }
            __syncthreads();
            // prefetch next chunk into L2/WGP while this chunk computes
            if (kc + KC < DIMD) {
                // X next: 16KB = 256 lanes * 64B
                __builtin_prefetch(x + ((long)(b * SEQL + s0 + (t >> 2))) * DIMD + (kc + KC) + (t & 3) * 16, 0, 1);
                // W1 next: 32KB = 256 lanes * 128B
                __builtin_prefetch(W1e + (long)(kc + KC + (t >> 2)) * SUBH + (t & 3) * 32, 0, 1);
            }
            for (int k4 = 0; k4 < KC; k4 += 4) {
                // A frag: lane lm -> row, K pair (k4+2*lh, +1): one ds_load_b64
                v2f a = *(const v2f*)(lds + XS_OFF + (strip * 16 + lm) * XS_STR + k4 + 2 * lh);
                const int kb = k4 + 2 * lh;
#pragma unroll
                for (int n = 0; n < 4; ++n) {
                    const int col = half * 64 + n * 16 + lm;
                    v2f bf;
                    bf[0] = lds[W1_OFF + (kb + 0) * W1_STR + col];
                    bf[1] = lds[W1_OFF + (kb + 1) * W1_STR + col];
                    acc[n] = __builtin_amdgcn_wmma_f32_16x16x4_f32(
                        false, a, false, bf, (short)0, acc[n], false, false);
                }
            }
        }
        // epilogue: + b1 (from LDS), exact GELU (matches approximate=False),
        // write H (disjoint region; barrier at loop top orders reuse)
#pragma unroll
        for (int n = 0; n < 4; ++n) {
            const int col = half * 64 + n * 16 + lm;
            const float bias = lds[B1_OFF + slot * SUBH + col];
#pragma unroll
            for (int v = 0; v < 8; ++v) {
                const int M = strip * 16 + v + 8 * lh;
                float hv = acc[n][v] + bias;
                hv = 0.5f * hv * (1.0f + erff(hv * 0.70710678118654752f));
                H[M * H_STR + col] = hv;
            }
        }
    }

    // ---------------- Phase B: OUT = H0 @ W2[e0] + H1 @ W2[e1] + bsum ------
    for (int ng = 0; ng < 4; ++ng) {          // 128-col output groups
        v8f c[4] = {};
        for (int slot = 0; slot < 2; ++slot) {
            const int e = slot ? e1 : e0;
            const float* W2e = W2 + (long)e * SUBH * DIMD;
            const float* H = lds + (slot ? H1_OFF : H0_OFF);
            __syncthreads();
            // stage W2 chunk [128 k][128 col], stride 132 (overlays phase-A region)
            for (int i = t; i < SUBH * 128 / 4; i += 256) {  // 4096 float4
                const int k = i >> 5, q = i & 31;
                float4 v = *(const float4*)(W2e + (long)k * DIMD + ng * 128 + q * 4);
                *(float4*)(lds + W2_OFF + k * W2_STR + q * 4) = v;
            }
            __syncthreads();
            // prefetch the next (slot,ng) W2 chunk: 64KB = 256 lanes * 256B
            {
                const int nslot = slot ^ 1;
                const int nng = slot ? (ng + 1) : ng;
                if (nng < 4) {
                    const int ne_ = nslot ? e1 : e0;
                    const float* W2n = W2 + (long)ne_ * SUBH * DIMD + (long)(t >> 1) * DIMD + nng * 128;
                    __builtin_prefetch(W2n + (t & 1) * 64, 0, 1);
                }
            }
            for (int k4 = 0; k4 < SUBH; k4 += 4) {
                v2f a = *(const v2f*)(H + (strip * 16 + lm) * H_STR + k4 + 2 * lh);
                const int kb = k4 + 2 * lh;
#pragma unroll
                for (int n = 0; n < 4; ++n) {
                    const int col = half * 64 + n * 16 + lm;
                    v2f bf;
                    bf[0] = lds[W2_OFF + (kb + 0) * W2_STR + col];
                    bf[1] = lds[W2_OFF + (kb + 1) * W2_STR + col];
                    c[n] = __builtin_amdgcn_wmma_f32_16x16x4_f32(
                        false, a, false, bf, (short)0, c[n], false, false);
                }
            }
        }
        // store (bsum region never rewritten -> no race with next ng's staging)
#pragma unroll
        for (int n = 0; n < 4; ++n) {
            const int col = ng * 128 + half * 64 + n * 16 + lm;
            const float bias = lds[BS_OFF + col];
#pragma unroll
            for (int v = 0; v < 8; ++v) {
                const int M = strip * 16 + v + 8 * lh;
                out[((long)(b * SEQL + s0 + M)) * DIMD + col] = c[n][v] + bias;
            }
        }
    }
}

// ---------------------------------------------------------------------------
extern "C" void kernel_launch(void* const* d_in, const int* in_sizes, int n_in,
                              void* d_out, int out_size, void* d_ws, size_t ws_size,
                              hipStream_t stream)
{
    const float* x   = (const float*)d_in[0];
    const float* Wsw = (const float*)d_in[1];
    const float* bsw = (const float*)d_in[2];
    const float* W1  = (const float*)d_in[3];
    const float* b1  = (const float*)d_in[4];
    const float* W2  = (const float*)d_in[5];
    const float* b2  = (const float*)d_in[6];
    float* out = (float*)d_out;

    // workspace: [NB][RCHUNKS][NE] f32 partials (128 KB) + top2 int[NB*2]
    float* partials = (float*)d_ws;
    int*   top2     = (int*)((char*)d_ws + (size_t)NB * RCHUNKS * NE * sizeof(float));

    router_partial<<<dim3(RCHUNKS, NB), 256, 0, stream>>>(x, Wsw, partials);
    reduce_top2<<<1, 256, 0, stream>>>(partials, bsw, top2);

    const size_t shmem = (size_t)LDS_FLOATS * sizeof(float);   // 137,216 B (<320KB WGP LDS)
    (void)hipFuncSetAttribute((const void*)moe_ffn,
                              hipFuncAttributeMaxDynamicSharedMemorySize, (int)shmem);
    moe_ffn<<<dim3(SEQL / TS, NB), 256, shmem, stream>>>(x, W1, b1, W2, b2, top2, out);
}